// DualAttnModel_816043786693
// MI455X (gfx1250) — compile-verified
//
#include <hip/hip_runtime.h>

typedef __attribute__((ext_vector_type(2))) float v2f;
typedef __attribute__((ext_vector_type(8))) float v8f;

#define NB 32
#define NP 16
#define NL 512
#define ND 256

// ---------------------------------------------------------------------------
// Kernel 1: one block per (b,p) page. 256 threads = 8 waves (wave32).
//   phase 0: gather token row offsets into LDS
//   phase 1: word scores = (H . V)/10  (coalesced float4 loads, wave reduce)
//   phase 2: masked softmax over L=512
//   phase 3: webpage_vec = attn^T . H   via V_WMMA_F32_16X16X4_F32
//   phase 4: page score = (webpage_vec . W)/10 (+ page mask)
// ---------------------------------------------------------------------------
__global__ __launch_bounds__(256) void dualattn_page_kernel(
    const int*   __restrict__ seq_ids,
    const int*   __restrict__ num_pages,
    const int*   __restrict__ seq_lengths,
    const float* __restrict__ emb,
    const float* __restrict__ V,
    const float* __restrict__ W,
    float* __restrict__ out_attn,
    float* __restrict__ out_pscore,
    float* __restrict__ out_wvec)
{
    __shared__ int   s_off[NL];   // token row offsets (elements) into emb
    __shared__ float s_sc[NL];    // scores, then attn weights
    __shared__ float s_y[ND];     // webpage_vec for this page
    __shared__ float s_red[8];

    const int bp   = blockIdx.x;      // b*NP + p
    const int b    = bp >> 4;
    const int p    = bp & 15;
    const int tid  = threadIdx.x;
    const int lane = tid & 31;
    const int wid  = tid >> 5;        // wave id 0..7

    // ---- phase 0: token offsets ----
    for (int l = tid; l < NL; l += 256)
        s_off[l] = seq_ids[bp * NL + l] * ND;

    const int  npages = num_pages[b];
    const int  slen   = seq_lengths[bp];
    const bool pv     = (p < npages);
    __syncthreads();

    // ---- phase 1: word scores ----
    const float4* V4 = reinterpret_cast<const float4*>(V);
    float4 v0 = V4[lane];        // d = 4*lane .. 4*lane+3
    float4 v1 = V4[lane + 32];   // d = 128 + 4*lane ..

    for (int l = wid * 64; l < wid * 64 + 64; ++l) {
        const float4* row = reinterpret_cast<const float4*>(emb + s_off[l]);
        float4 h0 = row[lane];
        float4 h1 = row[lane + 32];
        float acc = h0.x*v0.x + h0.y*v0.y + h0.z*v0.z + h0.w*v0.w
                  + h1.x*v1.x + h1.y*v1.y + h1.z*v1.z + h1.w*v1.w;
        #pragma unroll
        for (int o = 16; o > 0; o >>= 1) acc += __shfl_xor(acc, o, 32);
        if (lane == 0) {
            float sc = acc / 10.0f;
            if (pv && l >= slen) sc = -999.0f;   // ragged word mask
            s_sc[l] = sc;
        }
    }
    __syncthreads();

    // ---- phase 2: softmax over 512 (2 elems/thread) ----
    float x0 = s_sc[tid], x1 = s_sc[tid + 256];
    float m = fmaxf(x0, x1);
    #pragma unroll
    for (int o = 16; o > 0; o >>= 1) m = fmaxf(m, __shfl_xor(m, o, 32));
    if (lane == 0) s_red[wid] = m;
    __syncthreads();
    float mall = s_red[0];
    #pragma unroll
    for (int i = 1; i < 8; ++i) mall = fmaxf(mall, s_red[i]);
    float e0 = expf(x0 - mall), e1 = expf(x1 - mall);
    float s = e0 + e1;
    #pragma unroll
    for (int o = 16; o > 0; o >>= 1) s += __shfl_xor(s, o, 32);
    __syncthreads();
    if (lane == 0) s_red[wid] = s;
    __syncthreads();
    float sall = 0.f;
    #pragma unroll
    for (int i = 0; i < 8; ++i) sall += s_red[i];
    float inv = 1.0f / sall;
    float a0 = e0 * inv, a1 = e1 * inv;
    s_sc[tid] = a0; s_sc[tid + 256] = a1;        // own slots only: no race
    out_attn[bp * NL + tid]       = a0;
    out_attn[bp * NL + tid + 256] = a1;
    __syncthreads();

    // ---- phase 3: webpage_vec[d] = sum_l attn[l]*H[l][d] via WMMA f32 16x16x4
    // A(16x4): lane m holds H[l0+k][d0+m]; lanes 0-15 -> K=0,1; 16-31 -> K=2,3.
    // B(4x16): attn[l0+k] broadcast across N (rows split lo/hi lane halves).
    // All 16 D columns identical -> lanes 0 and 16 carry the result column.
    const int half = lane >> 4;
    const int lm   = lane & 15;
    const int d0   = wid * 16;        // first d-chunk of this wave
    const int d1   = d0 + 128;        // second d-chunk
    v8f c0 = {0.f,0.f,0.f,0.f,0.f,0.f,0.f,0.f};
    v8f c1 = {0.f,0.f,0.f,0.f,0.f,0.f,0.f,0.f};
    #pragma unroll 4
    for (int l0 = 0; l0 < NL; l0 += 4) {
        const int k0 = l0 + 2 * half;
        const int r0 = s_off[k0];
        const int r1 = s_off[k0 + 1];
        v2f Bv; Bv[0] = s_sc[k0];            Bv[1] = s_sc[k0 + 1];
        v2f A0; A0[0] = emb[r0 + d0 + lm];   A0[1] = emb[r1 + d0 + lm];
        v2f A1; A1[0] = emb[r0 + d1 + lm];   A1[1] = emb[r1 + d1 + lm];
        c0 = __builtin_amdgcn_wmma_f32_16x16x4_f32(false, A0, false, Bv,
                                                   (short)0, c0, false, false);
        c1 = __builtin_amdgcn_wmma_f32_16x16x4_f32(false, A1, false, Bv,
                                                   (short)0, c1, false, false);
    }
    if (lm == 0) {   // lanes 0 and 16: column N=0 holds y (M=r / M=8+r per VGPR)
        #pragma unroll
        for (int r = 0; r < 8; ++r) {
            s_y[d0 + 8 * half + r] = c0[r];
            s_y[d1 + 8 * half + r] = c1[r];
        }
    }
    __syncthreads();

    // ---- phase 4: page score ----
    float yv = s_y[tid];
    out_wvec[bp * ND + tid] = yv;
    float ps = yv * W[tid];
    #pragma unroll
    for (int o = 16; o > 0; o >>= 1) ps += __shfl_xor(ps, o, 32);
    if (lane == 0) s_red[wid] = ps;
    __syncthreads();
    if (tid == 0) {
        float t = 0.f;
        for (int i = 0; i < 8; ++i) t += s_red[i];
        t /= 10.0f;
        if (p >= npages) t = -9999.0f;       // page mask (post-compute, as ref)
        out_pscore[bp] = t;
    }
}

// ---------------------------------------------------------------------------
// Kernel 2: one block per batch element. Sparsemax over P=16, final vector,
// decoder + sigmoid.
// ---------------------------------------------------------------------------
__global__ __launch_bounds__(256) void dualattn_head_kernel(
    const float* __restrict__ dec_w,
    const float* __restrict__ pscore,
    const float* __restrict__ wvec,
    float* __restrict__ out_probs,
    float* __restrict__ out_senti,
    float* __restrict__ out_pattn,
    float* __restrict__ out_fvec)
{
    __shared__ float s_pa[NP];
    __shared__ float s_red[8];
    const int b    = blockIdx.x;
    const int tid  = threadIdx.x;
    const int lane = tid & 31;
    const int wid  = tid >> 5;

    if (tid == 0) {
        float z[NP], zs[NP];
        for (int i = 0; i < NP; ++i) { z[i] = pscore[b * NP + i]; zs[i] = z[i]; }
        // insertion sort descending (16 elems, trivial)
        for (int i = 1; i < NP; ++i) {
            float key = zs[i]; int j = i - 1;
            while (j >= 0 && zs[j] < key) { zs[j + 1] = zs[j]; --j; }
            zs[j + 1] = key;
        }
        float cs[NP]; float run = 0.f;
        for (int k = 0; k < NP; ++k) { run += zs[k]; cs[k] = run; }
        int kmax = 0;
        for (int k = 1; k <= NP; ++k)
            if (1.0f + (float)k * zs[k - 1] > cs[k - 1]) ++kmax;
        float tau = (cs[kmax - 1] - 1.0f) / (float)kmax;
        for (int i = 0; i < NP; ++i) {
            float pa = fmaxf(z[i] - tau, 0.f);
            s_pa[i] = pa;
            out_pattn[b * NP + i] = pa;
        }
    }
    __syncthreads();

    float f = 0.f;
    #pragma unroll
    for (int pp = 0; pp < NP; ++pp)
        f += s_pa[pp] * wvec[(b * NP + pp) * ND + tid];
    out_fvec[b * ND + tid] = f;

    float sd = f * dec_w[tid];
    #pragma unroll
    for (int o = 16; o > 0; o >>= 1) sd += __shfl_xor(sd, o, 32);
    if (lane == 0) s_red[wid] = sd;
    __syncthreads();
    if (tid == 0) {
        float senti = 0.f;
        for (int i = 0; i < 8; ++i) senti += s_red[i];
        out_senti[b] = senti;
        out_probs[b] = 1.0f / (1.0f + expf(-senti));
    }
}

// ---------------------------------------------------------------------------
extern "C" void kernel_launch(void* const* d_in, const int* in_sizes, int n_in,
                              void* d_out, int out_size, void* d_ws, size_t ws_size,
                              hipStream_t stream) {
    const int*   seq_ids   = (const int*)d_in[0];
    const int*   num_pages = (const int*)d_in[1];
    const int*   seq_len   = (const int*)d_in[2];
    const float* emb       = (const float*)d_in[3];
    const float* V         = (const float*)d_in[4];
    const float* W         = (const float*)d_in[5];
    const float* dec_w     = (const float*)d_in[6];

    float* out = (float*)d_out;
    // flat output layout, reference return order
    float* o_probs  = out;                                // [32]
    float* o_senti  = out + 32;                           // [32]
    float* o_attn   = out + 64;                           // [32*16*512]
    float* o_pattn  = o_attn + NB * NP * NL;              // [32*16]
    float* o_fvec   = o_pattn + NB * NP;                  // [32*256]
    float* o_pscore = o_fvec + NB * ND;                   // [32*16]
    float* o_wvec   = o_pscore + NB * NP;                 // [32*16*256]

    dualattn_page_kernel<<<dim3(NB * NP), dim3(256), 0, stream>>>(
        seq_ids, num_pages, seq_len, emb, V, W, o_attn, o_pscore, o_wvec);
    dualattn_head_kernel<<<dim3(NB), dim3(256), 0, stream>>>(
        dec_w, o_pscore, o_wvec, o_probs, o_senti, o_pattn, o_fvec);
}